// Word2MatEncoder_43516608643660
// MI455X (gfx1250) — compile-verified
//
#include <hip/hip_runtime.h>

// ---------------------------------------------------------------------------
// Word2Mat encoder for MI455X (gfx1250).
//
//   out[b] = M(sent[b,0]) @ M(sent[b,1]) @ ... @ M(sent[b,127])   (28x28 f32)
//
// - fp32 WMMA (v_wmma_f32_16x16x4_f32) keeps reference fp32 semantics over
//   the 128-step product chain (bf16/f16 would compound error badly).
// - Rows 0-15 / 16-31 of the running product evolve independently -> 2 waves
//   per batch element = 1024 single-wave workgroups, zero barriers (DS ops
//   are in-order within a wave).
// - Per step: 16x32 x 32x32 = 4 partial accumulators x 4 K-chunks
//   (kb-split halves the WMMA dependency chain: 4 deep instead of 8).
// - Double-buffered global->LDS prefetch of the next word matrix overlaps
//   the gather latency with the WMMA block; table (94 MB) is L2-resident
//   (192 MB), so duplicate reads by the two row-halves are cheap.
// ---------------------------------------------------------------------------

typedef __attribute__((ext_vector_type(2))) float v2f;
typedef __attribute__((ext_vector_type(4))) float v4f;
typedef __attribute__((ext_vector_type(8))) float v8f;

#define BATCH 512
#define SEQ   128
#define MD    28
#define DIM   784     // 28*28
#define PAD   32      // padded matrix dim
#define NF4   196     // DIM/4 float4 chunks per word-matrix row

__global__ __launch_bounds__(32)
void w2m_chain_kernel(const int* __restrict__ sent,
                      const float* __restrict__ table,
                      float* __restrict__ out)
{
    // Per-wave LDS scratch. Workgroup == one wave32, so DS in-order execution
    // gives all the synchronization we need (no barriers).
    __shared__ float cur[16 * PAD];          // wave's 16 rows of running product (row-major, 32 cols)
    __shared__ float wbuf[2][PAD * PAD];     // double-buffered next word matrix (row-major, zero-padded)
    __shared__ int   idxs[SEQ];              // this batch element's token indices

    const int lane = threadIdx.x;            // 0..31
    const int bid  = blockIdx.x;             // 0..1023
    const int b    = bid >> 1;               // batch element
    const int half = bid & 1;                // 0: global rows 0-15, 1: global rows 16-31 (16-27 valid)

    const int ln = lane & 15;                // 0..15
    const int hh = lane >> 4;                // 0 or 1 (half-wave id per WMMA layouts)

    // ---- zero-init LDS so the 28->32 pad regions stay zero forever ----
    {
        v4f z4 = {0.f, 0.f, 0.f, 0.f};
        for (int i = lane; i < (16 * PAD) / 4; i += 32)      ((v4f*)cur)[i]          = z4;
        for (int i = lane; i < (2 * PAD * PAD) / 4; i += 32) ((v4f*)&wbuf[0][0])[i]  = z4;
    }

    // ---- preload all 128 token indices for this batch element ----
    {
        const int* srow = sent + b * SEQ;
        #pragma unroll
        for (int t = 0; t < SEQ / 32; ++t)
            idxs[lane + 32 * t] = srow[lane + 32 * t];
    }

    // ---- seed: rows [16*half, 16*half+16) of word matrix sent[b,0] -> cur ----
    {
        const float* r0 = table + (long)idxs[0] * DIM;
        const int fbase = half ? 112 : 0;    // float4 index of first row we own (row*7)
        const int fend  = half ? NF4 : 112;
        #pragma unroll
        for (int t = 0; t < 4; ++t) {
            int f  = fbase + lane + 32 * t;
            int fc = f < fend ? f : fend - 1;              // clamped load: EXEC stays full
            v4f v  = *(const v4f*)(r0 + fc * 4);
            if (f < fend) {                                // guarded store only
                int R = fc / 7, q = fc % 7;                // 7 float4 per 28-wide row
                *(v4f*)(cur + (R - 16 * half) * PAD + q * 4) = v;
            }
        }
    }

    // ---- stage word matrix sent[b,1] into wbuf[0] ----
    {
        const float* r1 = table + (long)idxs[1] * DIM;
        #pragma unroll
        for (int t = 0; t < 7; ++t) {
            int f  = lane + 32 * t;
            int fc = f < NF4 ? f : NF4 - 1;
            v4f v  = *(const v4f*)(r1 + fc * 4);
            if (f < NF4) {
                int R = fc / 7, q = fc % 7;
                *(v4f*)(&wbuf[0][0] + R * PAD + q * 4) = v;
            }
        }
    }

    v4f pf[7];
    const v8f zero8 = {0.f, 0.f, 0.f, 0.f, 0.f, 0.f, 0.f, 0.f};

    for (int s = 1; s < SEQ; ++s) {
        const float* wc = &wbuf[(s - 1) & 1][0];
        float*       wn = &wbuf[s & 1][0];

        // -- prefetch word matrix for step s+1 into registers (overlaps WMMAs) --
        const bool more = (s + 1) < SEQ;
        if (more) {
            const float* rn = table + (long)idxs[s + 1] * DIM;
            #pragma unroll
            for (int t = 0; t < 7; ++t) {
                int f  = lane + 32 * t;
                int fc = f < NF4 ? f : NF4 - 1;
                pf[t] = *(const v4f*)(rn + fc * 4);
            }
        }

        // -- cur(16x32) @ w(32x32): 2 output tiles, kb-split accumulators --
        v8f acc0a = zero8, acc0b = zero8;   // output cols  0..15 (kb=0 / kb=1)
        v8f acc1a = zero8, acc1b = zero8;   // output cols 16..31 (kb=0 / kb=1)
        #pragma unroll
        for (int c = 0; c < 4; ++c) {
            #pragma unroll
            for (int kb = 0; kb < 2; ++kb) {
                const int k0 = kb * 16 + c * 4;
                const int kk = k0 + 2 * hh;
                // A fragment (16x4 f32): lane ln holds A[ln][kk], A[ln][kk+1]
                v2f a = *(const v2f*)(cur + ln * PAD + kk);
                // B fragments (4x16 f32): lane holds B[kk][n], B[kk+1][n]
                v2f b0, b1;
                b0.x = wc[kk * PAD + ln];
                b0.y = wc[(kk + 1) * PAD + ln];
                b1.x = wc[kk * PAD + 16 + ln];
                b1.y = wc[(kk + 1) * PAD + 16 + ln];
                if (kb == 0) {
                    acc0a = __builtin_amdgcn_wmma_f32_16x16x4_f32(
                                false, a, false, b0, (short)0, acc0a, false, false);
                    acc1a = __builtin_amdgcn_wmma_f32_16x16x4_f32(
                                false, a, false, b1, (short)0, acc1a, false, false);
                } else {
                    acc0b = __builtin_amdgcn_wmma_f32_16x16x4_f32(
                                false, a, false, b0, (short)0, acc0b, false, false);
                    acc1b = __builtin_amdgcn_wmma_f32_16x16x4_f32(
                                false, a, false, b1, (short)0, acc1b, false, false);
                }
            }
        }

        // -- combine kb partials and write running product back
        //    (D layout: VGPR r -> rows r / r+8) --
        #pragma unroll
        for (int r = 0; r < 8; ++r) {
            cur[(r + 8 * hh) * PAD + ln]      = acc0a[r] + acc0b[r];
            cur[(r + 8 * hh) * PAD + 16 + ln] = acc1a[r] + acc1b[r];
        }

        // -- stage prefetched matrix into the other buffer for step s+1 --
        if (more) {
            #pragma unroll
            for (int t = 0; t < 7; ++t) {
                int f = lane + 32 * t;
                if (f < NF4) {
                    int R = f / 7, q = f % 7;
                    *(v4f*)(wn + R * PAD + q * 4) = pf[t];
                }
            }
        }
    }

    // ---- emit our 16 (or 12) valid rows: out[b*784 + R*28 + 4q] ----
    {
        float* outb = out + (long)b * DIM;
        const int fbase = half ? 112 : 0;
        const int fend  = half ? NF4 : 112;
        #pragma unroll
        for (int t = 0; t < 4; ++t) {
            int f = fbase + lane + 32 * t;
            if (f < fend) {
                int R = f / 7, q = f % 7;
                *(v4f*)(outb + f * 4) = *(const v4f*)(cur + (R - 16 * half) * PAD + q * 4);
            }
        }
    }
}

extern "C" void kernel_launch(void* const* d_in, const int* in_sizes, int n_in,
                              void* d_out, int out_size, void* d_ws, size_t ws_size,
                              hipStream_t stream) {
    const int*   sent  = (const int*)d_in[0];     // [512,128] int32
    const float* table = (const float*)d_in[1];   // [30001,784] f32
    float*       outp  = (float*)d_out;           // [512,784] f32
    (void)in_sizes; (void)n_in; (void)out_size; (void)d_ws; (void)ws_size;

    w2m_chain_kernel<<<dim3(BATCH * 2), dim3(32), 0, stream>>>(sent, table, outp);
}